// LightGCN_21732534518149
// MI455X (gfx1250) — compile-verified
//
#include <hip/hip_runtime.h>

#define DIM 64
#define EPB 256   // edges staged per block
#define TPB 256   // threads per block (8 wave32 waves)
#define NUM_LAYERS 3

// ---------------- CDNA5 async global->LDS helpers ----------------
typedef int v2i __attribute__((vector_size(8)));
typedef v2i __attribute__((address_space(1)))* g_v2i_ptr;   // global (device) ptr
typedef v2i __attribute__((address_space(3)))* l_v2i_ptr;   // LDS ptr

__device__ __forceinline__ void async_copy_b64(void* lds_dst, const void* gsrc) {
#if defined(__gfx1250__) && __has_builtin(__builtin_amdgcn_global_load_async_to_lds_b64)
    __builtin_amdgcn_global_load_async_to_lds_b64(
        (g_v2i_ptr)gsrc,
        (l_v2i_ptr)lds_dst,
        0, 0);
#else
    unsigned lds_off = (unsigned)(unsigned long long)
        (__attribute__((address_space(3))) char*)lds_dst;
    asm volatile("global_load_async_to_lds_b64 %0, %1, off"
                 :: "v"(lds_off), "v"((unsigned long long)gsrc)
                 : "memory");
#endif
}

__device__ __forceinline__ void wait_asynccnt0() {
#if defined(__gfx1250__) && __has_builtin(__builtin_amdgcn_s_wait_asynccnt)
    __builtin_amdgcn_s_wait_asynccnt(0);
#else
    asm volatile("s_wait_asynccnt 0" ::: "memory");
#endif
}

__device__ __forceinline__ void atomic_add_f32(float* p, float v) {
    (void)__hip_atomic_fetch_add(p, v, __ATOMIC_RELAXED, __HIP_MEMORY_SCOPE_AGENT);
}

// ---------------- kernels ----------------

// One symmetric SpMM layer: xout[v] += w*x[u]; xout[u] += w*x[v].
// Edge chunk is staged into LDS with CDNA5 async global->LDS copies.
// (Defined first so the disasm snippet shows the async + atomic lowering.)
__global__ __launch_bounds__(TPB) void spmm_edges(const long long* __restrict__ adj,
                                                  const float* __restrict__ dis,
                                                  const float* __restrict__ xin,
                                                  float* __restrict__ xout, int E) {
    __shared__ long long uv64[2 * EPB];  // raw int64 endpoints (async dest)
    __shared__ int   su[EPB];
    __shared__ int   sv[EPB];
    __shared__ float sw[EPB];

    const int t    = threadIdx.x;
    const int base = blockIdx.x * EPB;
    const int e    = base + t;

    if (e < E) {
        async_copy_b64(&uv64[t],        &adj[e]);                // u endpoint
        async_copy_b64(&uv64[EPB + t],  &adj[(size_t)E + e]);    // v endpoint
    } else {
        uv64[t] = 0;
        uv64[EPB + t] = 0;
    }
    wait_asynccnt0();        // per-wave ASYNCcnt drain
    __syncthreads();         // make LDS staging visible block-wide

    {
        int u = (int)uv64[t];
        int v = (int)uv64[EPB + t];
        float w = (e < E) ? dis[u] * dis[v] : 0.0f;  // D^-1/2 A D^-1/2 weight
        su[t] = u;
        sv[t] = v;
        sw[t] = w;
    }
    __syncthreads();

    const int cnt = min(EPB, E - base);
    const int g = t >> 6;         // edge sub-group 0..3
    const int d = t & (DIM - 1);  // feature dim 0..63 (coalesced 256B lines)
    for (int k = g; k < cnt; k += TPB / DIM) {
        const int u = su[k];
        const int v = sv[k];
        const float w = sw[k];
        const size_t uo = (size_t)u * DIM + d;
        const size_t vo = (size_t)v * DIM + d;
        const float xu = xin[uo];
        const float xv = xin[vo];
        atomic_add_f32(&xout[vo], w * xu);   // forward edge
        atomic_add_f32(&xout[uo], w * xv);   // reversed edge
    }
}

// deg[u] += 1, deg[v] += 1 for every stored (u,v); equals degree of the
// symmetrized graph used by gcn_norm.
__global__ __launch_bounds__(TPB) void deg_kernel(const long long* __restrict__ adj,
                                                  float* __restrict__ deg, int E) {
    int e = blockIdx.x * blockDim.x + threadIdx.x;
    if (e < E) {
        int u = (int)adj[e];
        int v = (int)adj[(size_t)E + e];
        atomic_add_f32(&deg[u], 1.0f);
        atomic_add_f32(&deg[v], 1.0f);
    }
}

__global__ __launch_bounds__(TPB) void inv_sqrt_kernel(float* __restrict__ dis, int n) {
    int i = blockIdx.x * blockDim.x + threadIdx.x;
    if (i < n) {
        float d = dis[i];
        dis[i] = (d > 0.0f) ? __frsqrt_rn(d) : 0.0f;
    }
}

__global__ __launch_bounds__(TPB) void init_kernel(const float4* __restrict__ x,
                                                   float4* __restrict__ xcur,
                                                   float4* __restrict__ acc, int n4) {
    int i = blockIdx.x * blockDim.x + threadIdx.x;
    if (i < n4) {
        float4 v = x[i];
        xcur[i] = v;
        acc[i]  = v;
    }
}

__global__ __launch_bounds__(TPB) void acc_kernel(const float4* __restrict__ xn,
                                                  float4* __restrict__ acc, int n4) {
    int i = blockIdx.x * blockDim.x + threadIdx.x;
    if (i < n4) {
        float4 a = acc[i];
        float4 b = xn[i];
        a.x += b.x; a.y += b.y; a.z += b.z; a.w += b.w;
        acc[i] = a;
    }
}

__global__ __launch_bounds__(TPB) void scale_kernel(float4* __restrict__ acc,
                                                    float s, int n4) {
    int i = blockIdx.x * blockDim.x + threadIdx.x;
    if (i < n4) {
        float4 a = acc[i];
        a.x *= s; a.y *= s; a.z *= s; a.w *= s;
        acc[i] = a;
    }
}

// ---------------- launch ----------------
extern "C" void kernel_launch(void* const* d_in, const int* in_sizes, int n_in,
                              void* d_out, int out_size, void* d_ws, size_t ws_size,
                              hipStream_t stream) {
    const float*     x   = (const float*)d_in[0];      // [N, 64] f32
    const long long* adj = (const long long*)d_in[1];  // [2, E] i64
    float* out = (float*)d_out;                        // [N, 64] f32 accumulator

    const int N = in_sizes[0] / DIM;
    const int E = in_sizes[1] / 2;
    const size_t N64 = (size_t)N * DIM;
    const int n4 = (int)(N64 / 4);

    // workspace layout: dis[N] | bufA[N*64] | bufB[N*64]  (~51.6 MB)
    char* ws = (char*)d_ws;
    float* dis = (float*)ws;
    size_t off = (((size_t)N * sizeof(float)) + 255) & ~(size_t)255;
    float* bufA = (float*)(ws + off);
    float* bufB = (float*)(ws + off + N64 * sizeof(float));

    auto cdiv = [](long long a, long long b) { return (int)((a + b - 1) / b); };

    // degrees -> d^-1/2
    (void)hipMemsetAsync(dis, 0, (size_t)N * sizeof(float), stream);
    deg_kernel<<<cdiv(E, TPB), TPB, 0, stream>>>(adj, dis, E);
    inv_sqrt_kernel<<<cdiv(N, TPB), TPB, 0, stream>>>(dis, N);

    // x0 snapshot: acc = x, cur = x
    init_kernel<<<cdiv(n4, TPB), TPB, 0, stream>>>((const float4*)x,
                                                   (float4*)bufA, (float4*)out, n4);

    float* cur = bufA;
    float* nxt = bufB;
    for (int l = 0; l < NUM_LAYERS; ++l) {
        (void)hipMemsetAsync(nxt, 0, N64 * sizeof(float), stream);
        spmm_edges<<<cdiv(E, EPB), TPB, 0, stream>>>(adj, dis, cur, nxt, E);
        acc_kernel<<<cdiv(n4, TPB), TPB, 0, stream>>>((const float4*)nxt,
                                                      (float4*)out, n4);
        float* tmp = cur; cur = nxt; nxt = tmp;
    }

    // mean over (NUM_LAYERS + 1) snapshots
    scale_kernel<<<cdiv(n4, TPB), TPB, 0, stream>>>((float4*)out,
                                                    1.0f / (float)(NUM_LAYERS + 1), n4);
}